// SerializedEmbedding_7121055777167
// MI455X (gfx1250) — compile-verified
//
#include <hip/hip_runtime.h>

// ---------------------------------------------------------------------------
// SerializedEmbedding == plain embedding gather: out[r, :] = weight[idx[r], :]
// (each index lies in exactly one shard; masked shards contribute zero).
//
// MI455X strategy: Tensor Data Mover in GATHER MODE. One TENSOR_LOAD_TO_LDS
// gathers 8 table rows (16-bit row indices packed in D# groups 2/3) into a
// 32 KB LDS tile; one normal-mode TENSOR_STORE_FROM_LDS (TH=NT so the
// write-once output stream does not evict the ~206 MB table from the 192 MB
// L2) streams the tile to the contiguous output. Data never touches VGPRs;
// waves only build SGPR descriptors and fence on TENSORcnt.
// Pure-bandwidth problem: 128 MB traffic -> ~5.6 us floor at 23.3 TB/s.
//
// Toolchain: clang-23 / therock-10.0 6-arg TDM builtins:
//   (v4u g0, v8i g1, v4i g2, v4i g3, v8i extra, i32 cpol)
// ---------------------------------------------------------------------------

typedef unsigned int v4u __attribute__((ext_vector_type(4)));
typedef int          v8i __attribute__((ext_vector_type(8)));
typedef int          v4i __attribute__((ext_vector_type(4)));

#define ROWS_PER_TILE 8           // 8 rows x 1024 f32 = 32 KB LDS tile
#define TDM_TYPE_IMAGE (2u << 30) // D# group0 bits 127:126 = 2 ("image")
#define TH_STORE_NT 1             // store temporal hint: non-temporal

__global__ __launch_bounds__(32)
void emb_gather_tdm_kernel(const int* __restrict__ indices,
                           const float* __restrict__ weight,
                           float* __restrict__ out,
                           int nrows,   // total indices (16384)
                           int D,       // embedding dim (1024)
                           int nemb)    // table rows (50304)
{
    __shared__ float lbuf[ROWS_PER_TILE * 1024];   // 32 KB

    const int tile = blockIdx.x;
    const int row0 = tile * ROWS_PER_TILE;
    int valid = nrows - row0;
    if (valid > ROWS_PER_TILE) valid = ROWS_PER_TILE;
    valid = __builtin_amdgcn_readfirstlane(valid);

    // ---- fetch this tile's indices uniformly (SGPRs) ---------------------
    // Full tiles: one aligned 8-dword uniform load (row0 is 8-aligned, so
    // indices+row0 is 32B-aligned). Tail tile: clamped scalar loads; slots
    // >= valid are ignored by TDM (tile_dim1 = valid) so contents are moot.
    unsigned u[ROWS_PER_TILE];
    if (valid == ROWS_PER_TILE) {
        v8i iv = *(const v8i*)(indices + row0);
#pragma unroll
        for (int i = 0; i < ROWS_PER_TILE; ++i)
            u[i] = (unsigned)__builtin_amdgcn_readfirstlane(iv[i]) & 0xFFFFu;
    } else {
        const int last = nrows - 1;
#pragma unroll
        for (int i = 0; i < ROWS_PER_TILE; ++i) {
            int r = row0 + i;
            if (r > last) r = last;            // branchless s_min clamp
            u[i] = (unsigned)__builtin_amdgcn_readfirstlane(indices[r]) & 0xFFFFu;
        }
    }

    // LDS byte offset = low 32 bits of the generic (aperture) address.
    const unsigned lds = (unsigned)(size_t)(void*)lbuf;
    const unsigned long long waddr = (unsigned long long)(size_t)weight;
    const unsigned long long oaddr =
        (unsigned long long)(size_t)(out + (size_t)row0 * (size_t)D);

    // =======================================================================
    // D# for TENSOR_LOAD_TO_LDS  (gather mode, 16-bit indices)
    // =======================================================================
    v4u g0l;
    g0l.x = 0x80000001u;                       // count=1 | gather_mode=1 | 16b idx
    g0l.y = lds;                               // lds_addr
    g0l.z = (unsigned)(waddr & 0xFFFFFFFFu);   // global_addr[31:0]
    g0l.w = ((unsigned)(waddr >> 32) & 0x01FFFFFFu) | TDM_TYPE_IMAGE;

    v8i g1l;
    g1l[0] = 0x00020000;                       // data_size = 2 (4 bytes/elem)
    g1l[1] = (D & 0xFFFF) << 16;               // tensor_dim0[15:0]
    g1l[2] = ((D >> 16) & 0xFFFF)              // tensor_dim0[31:16]
           | ((nemb & 0xFFFF) << 16);          // tensor_dim1[15:0]
    g1l[3] = ((nemb >> 16) & 0xFFFF)           // tensor_dim1[31:16]
           | ((D & 0xFFFF) << 16);             // tile_dim0 = D
    g1l[4] = valid & 0xFFFF;                   // tile_dim1 = #valid indices
    g1l[5] = D;                                // tensor_dim0_stride[31:0]
    g1l[6] = 0;                                // stride[47:32] | dim1_stride (ignored)
    g1l[7] = 0;

    v4i g2, g3;                                // gather row indices, 16-bit packed
    g2[0] = (int)(u[0] | (u[1] << 16));
    g2[1] = (int)(u[2] | (u[3] << 16));
    g2[2] = (int)(u[4] | (u[5] << 16));
    g2[3] = (int)(u[6] | (u[7] << 16));
    g3[0] = 0; g3[1] = 0; g3[2] = 0; g3[3] = 0;

    v8i gx;                                    // unused extra operand (VADDR4)
    gx[0] = 0; gx[1] = 0; gx[2] = 0; gx[3] = 0;
    gx[4] = 0; gx[5] = 0; gx[6] = 0; gx[7] = 0;

    __builtin_amdgcn_tensor_load_to_lds(g0l, g1l, g2, g3, gx, /*cpol=*/0);

    // Ensure the gather finished writing LDS before the store descriptor
    // reads it.
    __builtin_amdgcn_s_wait_tensorcnt(0);

    // =======================================================================
    // D# for TENSOR_STORE_FROM_LDS  (normal mode, contiguous 2D tile DxVALID)
    // =======================================================================
    v4u g0s;
    g0s.x = 0x00000001u;                       // count=1, normal mode
    g0s.y = lds;
    g0s.z = (unsigned)(oaddr & 0xFFFFFFFFu);
    g0s.w = ((unsigned)(oaddr >> 32) & 0x01FFFFFFu) | TDM_TYPE_IMAGE;

    v8i g1s;
    g1s[0] = 0x00020000;                       // data_size = 4 bytes
    g1s[1] = (D & 0xFFFF) << 16;               // tensor_dim0[15:0]
    g1s[2] = ((D >> 16) & 0xFFFF)
           | ((valid & 0xFFFF) << 16);         // tensor_dim1 = valid rows
    g1s[3] = ((D & 0xFFFF) << 16);             // tile_dim0 = D
    g1s[4] = valid & 0xFFFF;                   // tile_dim1 = valid rows
    g1s[5] = D;                                // row stride = D (contiguous)
    g1s[6] = 0;
    g1s[7] = 0;

    v4i gz;
    gz[0] = 0; gz[1] = 0; gz[2] = 0; gz[3] = 0;

    // TH=NT: write-once output must not evict the L2-resident table.
    __builtin_amdgcn_tensor_store_from_lds(g0s, g1s, gz, gz, gx,
                                           /*cpol=*/TH_STORE_NT);
    __builtin_amdgcn_s_wait_tensorcnt(0);      // (S_ENDPGM also waits-idle)
}

extern "C" void kernel_launch(void* const* d_in, const int* in_sizes, int n_in,
                              void* d_out, int out_size, void* d_ws, size_t ws_size,
                              hipStream_t stream)
{
    (void)n_in; (void)d_ws; (void)ws_size;
    const int*   indices = (const int*)d_in[0];   // harness: integer -> const int*
    const float* weight  = (const float*)d_in[1];
    float*       out     = (float*)d_out;

    const int nrows = in_sizes[0];                // 8*2048 = 16384
    const int D     = out_size / nrows;           // 1024
    const int nemb  = in_sizes[1] / D;            // 50304

    const int ntiles = (nrows + ROWS_PER_TILE - 1) / ROWS_PER_TILE; // 2048
    emb_gather_tdm_kernel<<<ntiles, 32, 0, stream>>>(indices, weight, out,
                                                     nrows, D, nemb);
}